// MessagePassing_76914274337178
// MI455X (gfx1250) — compile-verified
//
#include <hip/hip_runtime.h>

#define DFEAT 64
#define WAVESZ 32   // CDNA5 is wave32

// ---------------------------------------------------------------------------
// Zero an int array (histogram init)
__global__ void mp_zero_i32(int* __restrict__ p, int n) {
    int i = blockIdx.x * blockDim.x + threadIdx.x;
    if (i < n) p[i] = 0;
}

// Zero a float array (fallback path only)
__global__ void mp_zero_f32(float* __restrict__ p, long long n) {
    long long i = (long long)blockIdx.x * blockDim.x + threadIdx.x;
    if (i < n) p[i] = 0.0f;
}

// ---------------------------------------------------------------------------
// Pass 1: degree histogram over destination nodes. Int atomics (cheap at L2),
// edge stream read once -> non-temporal so it doesn't evict x from L2.
__global__ void mp_count(const int* __restrict__ dstIdx, int E,
                         int* __restrict__ cnt) {
    int e = blockIdx.x * blockDim.x + threadIdx.x;
    if (e < E) {
        int d = __builtin_nontemporal_load(dstIdx + e);
        atomicAdd(&cnt[d], 1);
    }
}

// ---------------------------------------------------------------------------
// Pass 2: exclusive prefix-sum of degrees. Single 1024-thread workgroup:
// per-thread serial chunk sum, 1024-wide Hillis-Steele scan in LDS (ds_* ops),
// then per-thread serial write-out of exclusive offsets. 100k elements -> tiny.
// `cnt` is rewritten in place to become the scatter cursor (== offsets copy).
__global__ void __launch_bounds__(1024)
mp_scan(int* __restrict__ cnt /* in: counts, out: cursor */,
        int* __restrict__ offsets /* [N+1] */, int N) {
    __shared__ int lds[1024];
    int tid = threadIdx.x;
    int chunk = (N + 1023) / 1024;
    int lo = tid * chunk; if (lo > N) lo = N;
    int hi = lo + chunk;  if (hi > N) hi = N;

    int s = 0;
    for (int i = lo; i < hi; ++i) s += cnt[i];
    lds[tid] = s;
    __syncthreads();

    // inclusive scan across 1024 threads
    for (int off = 1; off < 1024; off <<= 1) {
        int v = (tid >= off) ? lds[tid - off] : 0;
        __syncthreads();
        lds[tid] += v;
        __syncthreads();
    }
    int running = lds[tid] - s;  // exclusive base for this thread's chunk
    for (int i = lo; i < hi; ++i) {
        int c = cnt[i];          // read BEFORE aliased write below
        offsets[i] = running;
        cnt[i]     = running;    // becomes cursor for mp_scatter
        running += c;
    }
    if (tid == 1023) offsets[N] = lds[1023];  // total == E
}

// ---------------------------------------------------------------------------
// Pass 3: bucket the source ids into CSR order.
__global__ void mp_scatter(const int* __restrict__ dstIdx,
                           const int* __restrict__ srcIdx, int E,
                           int* __restrict__ cursor,
                           int* __restrict__ srcSorted) {
    int e = blockIdx.x * blockDim.x + threadIdx.x;
    if (e < E) {
        int d = __builtin_nontemporal_load(dstIdx + e);
        int s = __builtin_nontemporal_load(srcIdx + e);
        int pos = atomicAdd(&cursor[d], 1);
        __builtin_nontemporal_store(s, srcSorted + pos);
    }
}

// ---------------------------------------------------------------------------
// Pass 4: wave-per-node aggregation, wave32-native.
//  * Up to 32 edge ids fetched with ONE coalesced global_load_b32 (lane-parallel),
//    then broadcast per-iteration via __shfl (no memory latency chain).
//  * Half-wave split: lanes 0-15 process even edges, lanes 16-31 odd edges.
//    Each half-wave loads a full 256B feature row as float4 (global_load_b128),
//    so two gathered rows are in flight per wave per iteration.
//  * Next row prefetched one edge ahead (global_prefetch_b8).
//  * Cross-half partial sums merged with __shfl_xor(.,16); no float atomics.
__global__ void __launch_bounds__(256)
mp_aggregate(const float* __restrict__ x,
             const int* __restrict__ offsets,
             const int* __restrict__ srcSorted,
             float* __restrict__ out, int N) {
    int wave = threadIdx.x >> 5;          // 8 waves per 256-thread block
    int lane = threadIdx.x & (WAVESZ - 1);
    int half = lane >> 4;                 // 0: even edges, 1: odd edges
    int fl   = lane & 15;                 // float4 slot within the 64-f row
    int n = blockIdx.x * 8 + wave;
    if (n >= N) return;

    int beg = offsets[n];
    int end = offsets[n + 1];

    float ax = 0.0f, ay = 0.0f, az = 0.0f, aw = 0.0f;

    for (int base = beg; base < end; base += WAVESZ) {
        int m = end - base;
        if (m > WAVESZ) m = WAVESZ;
        // one coalesced load covers up to 32 edge ids
        int myId = (lane < m) ? __builtin_nontemporal_load(srcSorted + base + lane)
                              : 0;
        for (int j = half; j < m; j += 2) {
            int s = __shfl(myId, j, WAVESZ);
            int jn = j + 2 < m ? j + 2 : j;
            int sn = __shfl(myId, jn, WAVESZ);
            const float4* rowN = (const float4*)(x + (size_t)sn * DFEAT) + fl;
            __builtin_prefetch(rowN, 0, 3);            // global_prefetch_b8
            float4 v = ((const float4*)(x + (size_t)s * DFEAT))[fl]; // b128
            ax += v.x; ay += v.y; az += v.z; aw += v.w;
        }
    }

    // merge the two half-wave partials (wave32 cross-half xor shuffle)
    ax += __shfl_xor(ax, 16, WAVESZ);
    ay += __shfl_xor(ay, 16, WAVESZ);
    az += __shfl_xor(az, 16, WAVESZ);
    aw += __shfl_xor(aw, 16, WAVESZ);

    if (half == 0) {
        float4 r; r.x = ax; r.y = ay; r.z = az; r.w = aw;
        ((float4*)(out + (size_t)n * DFEAT))[fl] = r;  // global_store_b128
    }
}

// ---------------------------------------------------------------------------
// Fallback (only if workspace is too small for CSR scratch): classic
// wave-per-edge float-atomic scatter-add.
__global__ void __launch_bounds__(256)
mp_atomic_fallback(const float* __restrict__ x,
                   const int* __restrict__ dstIdx,
                   const int* __restrict__ srcIdx,
                   float* __restrict__ out, int E) {
    int e = blockIdx.x * 8 + (threadIdx.x >> 5);
    int lane = threadIdx.x & (WAVESZ - 1);
    if (e < E) {
        int d = dstIdx[e];
        int s = srcIdx[e];
        float2 v = ((const float2*)(x + (size_t)s * DFEAT))[lane];
        float* o = out + (size_t)d * DFEAT + 2 * lane;
        atomicAdd(o,     v.x);   // global_atomic_add_f32 (no return)
        atomicAdd(o + 1, v.y);
    }
}

// ---------------------------------------------------------------------------
extern "C" void kernel_launch(void* const* d_in, const int* in_sizes, int n_in,
                              void* d_out, int out_size, void* d_ws, size_t ws_size,
                              hipStream_t stream) {
    const float* x  = (const float*)d_in[0];
    const int*   ei = (const int*)d_in[1];   // [2, E] row-major: row0=dst, row1=src
    float*       out = (float*)d_out;

    const int N = in_sizes[0] / DFEAT;
    const int E = in_sizes[1] / 2;
    const int* dstIdx = ei;
    const int* srcIdx = ei + E;

    // Workspace layout: cursor[N] | offsets[N+1] | srcSorted[E]
    const size_t need = (size_t)N * sizeof(int)
                      + (size_t)(N + 1) * sizeof(int)
                      + (size_t)E * sizeof(int);

    if (ws_size >= need) {
        int* cursor    = (int*)d_ws;
        int* offsets   = cursor + N;
        int* srcSorted = offsets + (N + 1);

        mp_zero_i32<<<(N + 255) / 256, 256, 0, stream>>>(cursor, N);
        mp_count  <<<(E + 255) / 256, 256, 0, stream>>>(dstIdx, E, cursor);
        mp_scan   <<<1, 1024, 0, stream>>>(cursor, offsets, N);
        mp_scatter<<<(E + 255) / 256, 256, 0, stream>>>(dstIdx, srcIdx, E,
                                                        cursor, srcSorted);
        mp_aggregate<<<(N + 7) / 8, 256, 0, stream>>>(x, offsets, srcSorted,
                                                      out, N);
    } else {
        long long total = (long long)N * DFEAT;
        mp_zero_f32<<<(int)((total + 255) / 256), 256, 0, stream>>>(out, total);
        mp_atomic_fallback<<<(E + 7) / 8, 256, 0, stream>>>(x, dstIdx, srcIdx,
                                                            out, E);
    }
}